// WaveFTLinear_58282706207416
// MI455X (gfx1250) — compile-verified
//
#include <hip/hip_runtime.h>

typedef __attribute__((ext_vector_type(2))) float v2f;
typedef __attribute__((ext_vector_type(4))) float v4f;
typedef __attribute__((ext_vector_type(8))) float v8f;
typedef __attribute__((ext_vector_type(4))) int   v4i;

typedef __attribute__((address_space(1))) v4i* gv4i_p;   // global int4*
typedef __attribute__((address_space(3))) v4i* lv4i_p;   // LDS int4*

#define M_TOTAL 8192
#define N_TOTAL 16384
#define K_TOTAL 4096
#define MT 128
#define NT 128
#define KT 32
#define NKT (K_TOTAL / KT)
#define LDSS 36   // LDS row stride in floats (16B-aligned rows, bank-conflict free)

// ---------------------------------------------------------------------------
// Kernel 1: W' = base_weight   (float4 copy, 67.1M floats)
// ---------------------------------------------------------------------------
__global__ __launch_bounds__(256) void wft_copy(const float* __restrict__ src,
                                                float* __restrict__ dst) {
  size_t i = (size_t)blockIdx.x * blockDim.x + threadIdx.x;   // one float4 each
  v4f v = *(const v4f*)(src + i * 4);
  *(v4f*)(dst + i * 4) = v;
}

// ---------------------------------------------------------------------------
// Kernel 2: W' += delta  (sparse separable IDWT patches, <=4x4 per coeff)
// ---------------------------------------------------------------------------
__constant__ float REC_LO[4] = { 0.48296291314469025f,  0.8365163037378079f,
                                 0.22414386804185735f, -0.12940952255092145f };
__constant__ float REC_HI[4] = {-0.12940952255092145f, -0.22414386804185735f,
                                 0.8365163037378079f,  -0.48296291314469025f };

__global__ __launch_bounds__(256) void wft_scatter(const float* __restrict__ spectrum,
                                                   const int* __restrict__ indices,
                                                   int nf,
                                                   float* __restrict__ W) {
  int t = blockIdx.x * blockDim.x + threadIdx.x;
  if (t >= nf) return;
  int i0 = indices[t];          // row index into (16384, 4096) weight grid
  int i1 = indices[nf + t];     // col index
  float s = spectrum[t] * 150.0f;

  // dense coeff at (r, c) = (i0+1, i1+1) in padded 16386 x 4098 grid
  bool top  = (i0 <= 8191);     // r < H2 = 8193
  bool left = (i1 <= 2047);     // c < W2 = 2049
  int ir = top  ? (i0 + 1) : (i0 - 8192);   // quadrant-local row
  int ic = left ? (i1 + 1) : (i1 - 2048);   // quadrant-local col

  // axis-0 (rows, OUT dim) filter chosen by column half; axis-1 by row half
  const float* fr = left ? REC_LO : REC_HI;
  const float* fc = top  ? REC_LO : REC_HI;

  int jr0 = 2 * ir - 2;
  int jc0 = 2 * ic - 2;
  #pragma unroll
  for (int a = 0; a < 4; ++a) {
    int jr = jr0 + a;
    if ((unsigned)jr >= (unsigned)N_TOTAL) continue;
    float wr = s * fr[a];
    #pragma unroll
    for (int b = 0; b < 4; ++b) {
      int jc = jc0 + b;
      if ((unsigned)jc >= (unsigned)K_TOTAL) continue;
      atomicAdd(&W[(size_t)jr * K_TOTAL + jc], wr * fc[b]);
    }
  }
}

// ---------------------------------------------------------------------------
// Async Global -> LDS staging (CDNA5 GLOBAL_LOAD_ASYNC_TO_LDS_B128, ASYNCcnt)
// ---------------------------------------------------------------------------
template <int N>
__device__ __forceinline__ void async_wait() {
#if __has_builtin(__builtin_amdgcn_s_wait_asynccnt)
  __builtin_amdgcn_s_wait_asynccnt(N);
#else
  asm volatile("s_wait_asynccnt %0" :: "i"(N) : "memory");
#endif
}

__device__ __forceinline__ void async_stage_tile(const float* __restrict__ gbase,
                                                 int grow0, int kb,
                                                 float* lds_tile, int tid) {
  #pragma unroll
  for (int i = 0; i < 4; ++i) {
    int id  = tid + i * 256;
    int row = id >> 3;                     // 8 float4 per 32-float row
    int kq  = (id & 7) * 4;
    const float* gp = gbase + (size_t)(grow0 + row) * K_TOTAL + kb + kq;
    float*       lp = lds_tile + row * LDSS + kq;          // 16B aligned (144B rows)
#if __has_builtin(__builtin_amdgcn_global_load_async_to_lds_b128)
    __builtin_amdgcn_global_load_async_to_lds_b128(
        (gv4i_p)(void*)const_cast<float*>(gp), (lv4i_p)(void*)lp, 0, 0);
#else
    unsigned loff = (unsigned)(size_t)lp;  // low 32 bits of shared aperture = LDS offset
    asm volatile("global_load_async_to_lds_b128 %0, %1, %2 offset:0"
                 :: "v"(loff), "v"((unsigned)0), "s"(gp) : "memory");
#endif
  }
}

// ---------------------------------------------------------------------------
// Kernel 3: out = X @ W'.T + bias   via V_WMMA_F32_16X16X4_F32
//   block = 256 threads (8 wave32), tile 128x128, double-buffered K-tile 32
//   async-to-LDS staging; wave (2x4) owns 64x32 = 4x2 WMMA accumulators
//   grid: blockIdx.x = M tiles (fast) so X stays L2-resident, W' streams once
// ---------------------------------------------------------------------------
__global__ __launch_bounds__(256) void wft_gemm(const float* __restrict__ X,
                                                const float* __restrict__ W,
                                                const float* __restrict__ bias,
                                                float* __restrict__ C) {
  __shared__ float sX[2][MT * LDSS];
  __shared__ float sW[2][NT * LDSS];

  const int tid   = threadIdx.x;
  const int lane  = tid & 31;
  const int wave  = tid >> 5;
  const int waveM = wave >> 2;   // 0..1  -> 64 rows each
  const int waveN = wave & 3;    // 0..3  -> 32 cols each
  const int m16   = lane & 15;
  const int g     = lane >> 4;   // half-wave select (K sub-stripe)

  const int m0 = blockIdx.x * MT;   // M on x: consecutive blocks share W' tile
  const int n0 = blockIdx.y * NT;

  v8f acc[4][2];
  #pragma unroll
  for (int ni = 0; ni < 2; ++ni) {
    float bv = bias[n0 + waveN * 32 + ni * 16 + m16];
    #pragma unroll
    for (int mi = 0; mi < 4; ++mi)
      #pragma unroll
      for (int v = 0; v < 8; ++v) acc[mi][ni][v] = bv;
  }

  // preload K-tile 0 into buffer 0
  async_stage_tile(X, m0, 0, sX[0], tid);
  async_stage_tile(W, n0, 0, sW[0], tid);

  for (int kt = 0; kt < NKT; ++kt) {
    const int cur = kt & 1;
    if (kt + 1 < NKT) {
      async_stage_tile(X, m0, (kt + 1) * KT, sX[cur ^ 1], tid);
      async_stage_tile(W, n0, (kt + 1) * KT, sW[cur ^ 1], tid);
      async_wait<8>();          // drain tile kt's 8 ops; tile kt+1 stays in flight
    } else {
      async_wait<0>();
    }
    __syncthreads();

    const float* bX = sX[cur];
    const float* bW = sW[cur];
    #pragma unroll
    for (int kk = 0; kk < KT; kk += 4) {
      v2f a[4], b[2];
      // A frag: lane holds X[m = lane%16][k = kk + 2g + {0,1}]
      #pragma unroll
      for (int mi = 0; mi < 4; ++mi)
        a[mi] = *(const v2f*)&bX[(waveM * 64 + mi * 16 + m16) * LDSS + kk + 2 * g];
      // B frag: lane holds W'[n = lane%16][k = kk + 2g + {0,1}]
      #pragma unroll
      for (int ni = 0; ni < 2; ++ni)
        b[ni] = *(const v2f*)&bW[(waveN * 32 + ni * 16 + m16) * LDSS + kk + 2 * g];

      #pragma unroll
      for (int mi = 0; mi < 4; ++mi)
        #pragma unroll
        for (int ni = 0; ni < 2; ++ni)
          acc[mi][ni] = __builtin_amdgcn_wmma_f32_16x16x4_f32(
              false, a[mi], false, b[ni], (short)0, acc[mi][ni], false, false);
    }
    __syncthreads();   // all reads of buf[cur] done before tile kt+2 overwrites it
  }

  // D layout: VGPR v, lane L -> row m = v + 8*(L/16), col n = L%16
  #pragma unroll
  for (int mi = 0; mi < 4; ++mi) {
    #pragma unroll
    for (int ni = 0; ni < 2; ++ni) {
      int col = n0 + waveN * 32 + ni * 16 + m16;
      #pragma unroll
      for (int v = 0; v < 8; ++v) {
        int row = m0 + waveM * 64 + mi * 16 + g * 8 + v;
        __builtin_nontemporal_store(acc[mi][ni][v], &C[(size_t)row * N_TOTAL + col]);
      }
    }
  }
}

// ---------------------------------------------------------------------------
extern "C" void kernel_launch(void* const* d_in, const int* in_sizes, int n_in,
                              void* d_out, int out_size, void* d_ws, size_t ws_size,
                              hipStream_t stream) {
  const float* x        = (const float*)d_in[0];   // (4,2048,4096) -> (8192,4096)
  const float* base_w   = (const float*)d_in[1];   // (16384,4096)
  const float* base_b   = (const float*)d_in[2];   // (16384,)
  const float* spectrum = (const float*)d_in[3];   // (NF,)
  const int*   indices  = (const int*)d_in[4];     // (2,NF)
  float*       out      = (float*)d_out;
  float*       Wp       = (float*)d_ws;            // 16384*4096*4 = 256 MiB

  const int nf = in_sizes[3];

  // W' = base_weight
  const size_t n_f4 = (size_t)N_TOTAL * K_TOTAL / 4;          // 16,777,216
  wft_copy<<<dim3((unsigned)(n_f4 / 256)), dim3(256), 0, stream>>>(base_w, Wp);

  // W' += 150 * IDWT(spectrum)  (sparse 4x4 patches)
  wft_scatter<<<dim3((nf + 255) / 256), dim3(256), 0, stream>>>(spectrum, indices, nf, Wp);

  // out = X @ W'.T + bias   (M tiles on x for L2-friendly rasterization)
  wft_gemm<<<dim3(M_TOTAL / MT, N_TOTAL / NT), dim3(256), 0, stream>>>(x, Wp, base_b, out);
}